// RIOUloss_46265387712856
// MI455X (gfx1250) — compile-verified
//
#include <hip/hip_runtime.h>

#define BLK 256

#if defined(__gfx1250__) && __has_builtin(__builtin_amdgcn_global_load_async_to_lds_b64)
#define HAVE_ASYNC_LDS 1
#else
#define HAVE_ASYNC_LDS 0
#endif

#if HAVE_ASYNC_LDS
typedef int v2i __attribute__((ext_vector_type(2)));
typedef __attribute__((address_space(1))) v2i* gas64_t;   // global (AS1) v2i*
typedef __attribute__((address_space(3))) v2i* las64_t;   // LDS    (AS3) v2i*
#endif

// Per-thread polygon slots: [0..6] = buffer A (cap 7), [7..12] = buffer B (cap 6),
// [13] = trash, [14] = pad. Stride 15 float2 = 30 dwords -> conflict-free wave32.
#define SLOTS 15
#define BUF_A 0
#define BUF_B 7
#define TRASH 13

__device__ __forceinline__ float side_eval(float edx, float edy, float cc,
                                           float x, float y) {
  // edx*y - edy*x - cc, 2 fmas
  return __builtin_fmaf(edx, y, -__builtin_fmaf(edy, x, cc));
}

// Clip polygon in LDS buffer RB (count cnt, <= MAXV) against half-plane, write to WB.
template <int RB, int WB, int MAXV>
__device__ __forceinline__ int clip_stage_lds(float2* base, int cnt,
                                              float ex, float ey,
                                              float edx, float edy) {
  const float cc = __builtin_fmaf(edx, ey, -(edy * ex));
  int m = 0;
  float2 pv = base[RB + ((cnt > 0) ? (cnt - 1) : 0)];     // one dynamic DS read
  float sprev = side_eval(edx, edy, cc, pv.x, pv.y);      // >=0 => inside (left)
#pragma unroll
  for (int v = 0; v < MAXV; ++v) {
    const float2 cvv = base[RB + v];                      // immediate-offset DS read
    const float scur = side_eval(edx, edy, cc, cvv.x, cvv.y);
    const bool act = (v < cnt);
    const bool inC = (scur >= 0.f);
    const bool inP = (sprev >= 0.f);
    const bool crossb = act && (inC != inP);
    const bool keep = act && inC;
    const float t = sprev * __builtin_amdgcn_rcpf(sprev - scur); // garbage if !crossb
    const float ixx = __builtin_fmaf(t, cvv.x - pv.x, pv.x);
    const float iyy = __builtin_fmaf(t, cvv.y - pv.y, pv.y);
    base[crossb ? (WB + m) : TRASH] = make_float2(ixx, iyy);     // unconditional store
    m += crossb ? 1 : 0;
    base[keep ? (WB + m) : TRASH] = cvv;                         // unconditional store
    m += keep ? 1 : 0;
    pv.x = act ? cvv.x : pv.x;
    pv.y = act ? cvv.y : pv.y;
    sprev = act ? scur : sprev;
  }
  return m;
}

__global__ __launch_bounds__(BLK) void riou_kernel(const float* __restrict__ pred,
                                                   const float* __restrict__ target,
                                                   float* __restrict__ out, int n) {
  __shared__ float2 sp[BLK * 3];
  __shared__ float2 st[BLK * 3];
  __shared__ float2 poly[BLK * SLOTS];

  const int tid = threadIdx.x;
  const int blockBase = blockIdx.x * BLK;
  const float2* gp = (const float2*)pred;
  const float2* gt = (const float2*)target;
  const int base3 = blockBase * 3;                 // fits in int (3M max)

  if (blockBase + BLK <= n) {                      // uniform fast path, no guards
#pragma unroll
    for (int k = 0; k < 3; ++k) {
      const int e = base3 + k * BLK + tid;
#if HAVE_ASYNC_LDS
      __builtin_amdgcn_global_load_async_to_lds_b64(
          (gas64_t)(gp + e), (las64_t)(&sp[k * BLK + tid]), 0, 0);
      __builtin_amdgcn_global_load_async_to_lds_b64(
          (gas64_t)(gt + e), (las64_t)(&st[k * BLK + tid]), 0, 0);
#else
      sp[k * BLK + tid] = gp[e];
      st[k * BLK + tid] = gt[e];
#endif
    }
  } else {                                         // tail block only
    const int totalElems = n * 3;
#pragma unroll
    for (int k = 0; k < 3; ++k) {
      const int e = base3 + k * BLK + tid;
      if (e < totalElems) {
#if HAVE_ASYNC_LDS
        __builtin_amdgcn_global_load_async_to_lds_b64(
            (gas64_t)(gp + e), (las64_t)(&sp[k * BLK + tid]), 0, 0);
        __builtin_amdgcn_global_load_async_to_lds_b64(
            (gas64_t)(gt + e), (las64_t)(&st[k * BLK + tid]), 0, 0);
#else
        sp[k * BLK + tid] = gp[e];
        st[k * BLK + tid] = gt[e];
#endif
      }
    }
  }
#if HAVE_ASYNC_LDS
#if __has_builtin(__builtin_amdgcn_s_wait_asynccnt)
  __builtin_amdgcn_s_wait_asynccnt(0);
#else
  asm volatile("s_wait_asynccnt 0" ::: "memory");
#endif
#endif
  __syncthreads();

  const int i = blockBase + tid;
  const bool valid = (i < n);

  // columns: [x, y, w, h, s, c];  rad = atan2(col4, col5)
  const float2 pxy = sp[tid * 3 + 0];
  const float2 pwh = sp[tid * 3 + 1];
  const float2 psc = sp[tid * 3 + 2];
  const float2 qxy = st[tid * 3 + 0];
  const float2 qwh = st[tid * 3 + 1];
  const float2 qsc = st[tid * 3 + 2];

  // sin/cos without atan2: cos(atan2(s,c)) = c*rsqrt(s^2+c^2), etc.
  const float n1 = psc.x * psc.x + psc.y * psc.y;
  const float r1 = (n1 > 0.f) ? __builtin_amdgcn_rsqf(n1) : 0.f;
  const float sv1 = psc.x * r1;
  const float cv1 = (n1 > 0.f) ? psc.y * r1 : 1.f;

  const float n2 = qsc.x * qsc.x + qsc.y * qsc.y;
  const float r2 = (n2 > 0.f) ? __builtin_amdgcn_rsqf(n2) : 0.f;
  const float sv2 = qsc.x * r2;
  const float cv2 = (n2 > 0.f) ? qsc.y * r2 : 1.f;

  // parallelogram corners via half-axis vectors (CCW: ++, -+, --, +-)
  float ax[4], ay[4], bx[4], by[4];
  {
    const float ux = 0.5f * pwh.x * cv1, uy = 0.5f * pwh.x * sv1;
    const float vx = -0.5f * pwh.y * sv1, vy = 0.5f * pwh.y * cv1;
    ax[0] = pxy.x + ux + vx; ay[0] = pxy.y + uy + vy;
    ax[1] = pxy.x - ux + vx; ay[1] = pxy.y - uy + vy;
    ax[2] = pxy.x - ux - vx; ay[2] = pxy.y - uy - vy;
    ax[3] = pxy.x + ux - vx; ay[3] = pxy.y + uy - vy;
    const float wx = 0.5f * qwh.x * cv2, wy = 0.5f * qwh.x * sv2;
    const float zx = -0.5f * qwh.y * sv2, zy = 0.5f * qwh.y * cv2;
    bx[0] = qxy.x + wx + zx; by[0] = qxy.y + wy + zy;
    bx[1] = qxy.x - wx + zx; by[1] = qxy.y - wy + zy;
    bx[2] = qxy.x - wx - zx; by[2] = qxy.y - wy - zy;
    bx[3] = qxy.x + wx - zx; by[3] = qxy.y + wy - zy;
  }

  float2* base = &poly[tid * SLOTS];

  // ---- Sutherland–Hodgman, stage 0: registers -> BUF_A (cnt fixed at 4) ----
  int cnt;
  {
    const float ex = bx[0], ey = by[0];
    const float edx = bx[1] - ex, edy = by[1] - ey;
    const float cc = __builtin_fmaf(edx, ey, -(edy * ex));
    int m = 0;
    float pvx = ax[3], pvy = ay[3];
    float sprev = side_eval(edx, edy, cc, pvx, pvy);
#pragma unroll
    for (int v = 0; v < 4; ++v) {
      const float cx = ax[v], cy = ay[v];
      const float scur = side_eval(edx, edy, cc, cx, cy);
      const bool inC = (scur >= 0.f);
      const bool inP = (sprev >= 0.f);
      const bool crossb = (inC != inP);
      const float t = sprev * __builtin_amdgcn_rcpf(sprev - scur);
      base[crossb ? (BUF_A + m) : TRASH] =
          make_float2(__builtin_fmaf(t, cx - pvx, pvx), __builtin_fmaf(t, cy - pvy, pvy));
      m += crossb ? 1 : 0;
      base[inC ? (BUF_A + m) : TRASH] = make_float2(cx, cy);
      m += inC ? 1 : 0;
      pvx = cx; pvy = cy; sprev = scur;
    }
    cnt = m;                                               // <= 5
  }

  // ---- stages 1, 2: LDS ping-pong ----
  cnt = clip_stage_lds<BUF_A, BUF_B, 5>(base, cnt, bx[1], by[1],
                                        bx[2] - bx[1], by[2] - by[1]);  // <= 6
  cnt = clip_stage_lds<BUF_B, BUF_A, 6>(base, cnt, bx[2], by[2],
                                        bx[3] - bx[2], by[3] - by[2]);  // <= 7

  // ---- stage 3 fused with shoelace: accumulate area from the emit stream ----
  float area2 = 0.f;
  {
    const float ex = bx[3], ey = by[3];
    const float edx = bx[0] - ex, edy = by[0] - ey;
    const float cc = __builtin_fmaf(edx, ey, -(edy * ex));
    float fx = 0.f, fy = 0.f, px = 0.f, py = 0.f;        // first / prev emitted
    int m = 0;
    float2 pv = base[BUF_A + ((cnt > 0) ? (cnt - 1) : 0)];
    float sprev = side_eval(edx, edy, cc, pv.x, pv.y);
#pragma unroll
    for (int v = 0; v < 7; ++v) {
      const float2 cvv = base[BUF_A + v];
      const float scur = side_eval(edx, edy, cc, cvv.x, cvv.y);
      const bool act = (v < cnt);
      const bool inC = (scur >= 0.f);
      const bool inP = (sprev >= 0.f);
      const bool crossb = act && (inC != inP);
      const bool keep = act && inC;
      const float t = sprev * __builtin_amdgcn_rcpf(sprev - scur);
      const float ixx = __builtin_fmaf(t, cvv.x - pv.x, pv.x);
      const float iyy = __builtin_fmaf(t, cvv.y - pv.y, pv.y);
      {  // emit intersection point
        const bool first = crossb && (m == 0);
        const float contrib = px * iyy - py * ixx;
        area2 += (crossb && !first) ? contrib : 0.f;
        fx = first ? ixx : fx;  fy = first ? iyy : fy;
        px = crossb ? ixx : px; py = crossb ? iyy : py;
        m += crossb ? 1 : 0;
      }
      {  // emit kept vertex
        const bool first = keep && (m == 0);
        const float contrib = px * cvv.y - py * cvv.x;
        area2 += (keep && !first) ? contrib : 0.f;
        fx = first ? cvv.x : fx; fy = first ? cvv.y : fy;
        px = keep ? cvv.x : px;  py = keep ? cvv.y : py;
        m += keep ? 1 : 0;
      }
      pv.x = act ? cvv.x : pv.x;
      pv.y = act ? cvv.y : pv.y;
      sprev = act ? scur : sprev;
    }
    area2 += px * fy - py * fx;   // wrap term; exactly 0 for m <= 2 (zero-init)
  }
  const float inter = 0.5f * fabsf(area2);

  const float uni = __builtin_fmaf(pwh.x, pwh.y, qwh.x * qwh.y) - inter;
  const float iou = inter * __builtin_amdgcn_rcpf(uni);

  // enclosing box over all 8 corners
  float minx = ax[0], maxx = ax[0], miny = ay[0], maxy = ay[0];
#pragma unroll
  for (int k = 1; k < 4; ++k) {
    minx = fminf(minx, ax[k]); maxx = fmaxf(maxx, ax[k]);
    miny = fminf(miny, ay[k]); maxy = fmaxf(maxy, ay[k]);
  }
#pragma unroll
  for (int k = 0; k < 4; ++k) {
    minx = fminf(minx, bx[k]); maxx = fmaxf(maxx, bx[k]);
    miny = fminf(miny, by[k]); maxy = fmaxf(maxy, by[k]);
  }
  const float wc = maxx - minx;
  const float hc = maxy - miny;
  const float c2diag = __builtin_fmaf(wc, wc, hc * hc);

  const float ddx = pxy.x - qxy.x;
  const float ddy = pxy.y - qxy.y;
  const float d2 = __builtin_fmaf(ddx, ddx, ddy * ddy);

  if (valid) out[i] = 1.f - iou + d2 * __builtin_amdgcn_rcpf(c2diag);
}

extern "C" void kernel_launch(void* const* d_in, const int* in_sizes, int n_in,
                              void* d_out, int out_size, void* d_ws, size_t ws_size,
                              hipStream_t stream) {
  const float* pred = (const float*)d_in[0];
  const float* target = (const float*)d_in[1];
  float* out = (float*)d_out;
  const int n = in_sizes[0] / 6;
  const int grid = (n + BLK - 1) / BLK;
  riou_kernel<<<grid, BLK, 0, stream>>>(pred, target, out, n);
}